// ReservoirComputer_19902878449925
// MI455X (gfx1250) — compile-verified
//
#include <hip/hip_runtime.h>
#include <math.h>

// Reservoir computer, MI455X (gfx1250, wave32, WMMA).
// h_{t+1} = 0.1*h_t + 0.9*tanh([x_t | h_t] @ [W_in; W]),  y_t = h_{t+1} @ r^T + b
// Persistent kernel: 16 WGs (1/WGP, == max cluster size) x 9 waves.
// Waves 0..7 of each WG: one 16x16 tile of h_new each (2 row-blocks x 64 col-tiles = 128 tiles).
// Wave 8 of WGs 0/1: readout tiles, pipelined one step behind.
// Per-step sync: cluster barrier (s_barrier_signal/wait -3) if clustered, else atomic barrier.
// Staging uses gfx1250 async global->LDS copies (ASYNCcnt) instead of VGPR round-trips.

#define B_    32
#define T_    4096
#define IN_   512
#define H_    1024
#define O_    16
#define KTOT  (IN_ + H_)     // 1536
#define KG_   (KTOT / 4)     // 384 k-groups of 4
#define KGX   (IN_ / 4)      // 128
#define KGH   (H_ / 4)       // 256
#define ALPHA 0.9f

#define NWG      16
#define NTHREADS 288         // 9 wave32 waves

// workspace layout (bytes); requires ws_size >= ~6.4 MB
#define WS_H   4096                        // h double buffer: 2*32*1024 f32 = 262144 B
#define WS_WP  (WS_H + 2 * B_ * H_ * 4)    // packed [W_in;W]: 1536*1024 f32 = 6 MB
#define WS_RP  (WS_WP + KTOT * H_ * 4)     // packed readout^T: 1024*16 f32 = 64 KB
#define LDS_BYTES (KG_ * 16 * 4 * 4)       // A slab: [384 kg][16 m][4 k] f32 = 96 KB

typedef float v2f __attribute__((ext_vector_type(2)));
typedef float v8f __attribute__((ext_vector_type(8)));
typedef int   v4i __attribute__((__vector_size__(16)));   // b128 payload type

#define AS1 __attribute__((address_space(1)))
#define AS3 __attribute__((address_space(3)))

// Async 16B global->LDS copy (GLOBAL_LOAD_ASYNC_TO_LDS_B128, tracked by ASYNCcnt).
// AS3 offset = low 32 bits of the generic LDS address.
__device__ __forceinline__ void async_copy16(const float* g, float* l) {
  __builtin_amdgcn_global_load_async_to_lds_b128(
      (AS1 v4i*)(uintptr_t)g,
      (AS3 v4i*)(unsigned)(uintptr_t)l,
      0, 0);
}
__device__ __forceinline__ void async_wait0() {
  asm volatile("s_wait_asynccnt 0x0" ::: "memory");
}

// ---------------- init / weight repack (stream-ordered before main) -------------
// Fragment-major pack: Wp[(kg*N + n)*4 + j] = Wfull[4*kg + j][n], so each lane's
// WMMA B fragment {B[k][n], B[k+1][n]} is one contiguous b64 load.
__global__ void rc_init(const float* __restrict__ Win, const float* __restrict__ W,
                        const float* __restrict__ Rw, float* __restrict__ ws_f,
                        unsigned* __restrict__ ws_u) {
  const size_t tid = (size_t)blockIdx.x * blockDim.x + threadIdx.x;
  const size_t nt  = (size_t)gridDim.x * blockDim.x;
  // zero barrier control + both h buffers
  for (size_t i = tid; i < WS_WP / 4; i += nt) ws_u[i] = 0u;
  // pack [W_in ; W]  (K=1536, N=1024)
  float* Wp = ws_f + WS_WP / 4;
  for (size_t idx = tid; idx < (size_t)KTOT * H_; idx += nt) {
    const int k = (int)(idx / H_), n = (int)(idx % H_);
    const float v = (k < IN_) ? Win[(size_t)k * H_ + n] : W[(size_t)(k - IN_) * H_ + n];
    Wp[((size_t)(k >> 2) * H_ + n) * 4 + (k & 3)] = v;
  }
  // pack readout^T (K=1024, N=16): B[k][n] = readout_w[n][k]
  float* Rp = ws_f + WS_RP / 4;
  for (size_t idx = tid; idx < (size_t)H_ * O_; idx += nt) {
    const int k = (int)(idx / O_), n = (int)(idx % O_);
    Rp[((size_t)(k >> 2) * O_ + n) * 4 + (k & 3)] = Rw[(size_t)n * H_ + k];
  }
}

// ---------------- grid-wide sync ------------------------------------------------
__device__ __forceinline__ void grid_sync(unsigned* cnt, unsigned* gen) {
  __threadfence();            // release: my h stores reach device scope
  __syncthreads();
  if (threadIdx.x == 0) {
    if (__builtin_amdgcn_cluster_id_x() != 0) {
      // dispatched as a 16-WG cluster: hardware split barrier (-3)
      __builtin_amdgcn_s_cluster_barrier();
    } else {
      // fallback: sense-reversing barrier over NWG co-resident workgroups
      unsigned g = __hip_atomic_load(gen, __ATOMIC_ACQUIRE, __HIP_MEMORY_SCOPE_AGENT);
      unsigned a = __hip_atomic_fetch_add(cnt, 1u, __ATOMIC_ACQ_REL, __HIP_MEMORY_SCOPE_AGENT);
      if (a == NWG - 1u) {
        __hip_atomic_store(cnt, 0u, __ATOMIC_RELAXED, __HIP_MEMORY_SCOPE_AGENT);
        __hip_atomic_fetch_add(gen, 1u, __ATOMIC_RELEASE, __HIP_MEMORY_SCOPE_AGENT);
      } else {
        while (__hip_atomic_load(gen, __ATOMIC_ACQUIRE, __HIP_MEMORY_SCOPE_AGENT) == g)
          __builtin_amdgcn_s_sleep(1);
      }
    }
  }
  __syncthreads();
  __threadfence();            // acquire: drop stale WGP$ lines before reading h
}

// ---------------- persistent scan kernel ---------------------------------------
__global__ void __launch_bounds__(NTHREADS)
rc_main(const float* __restrict__ x, const float* __restrict__ rbias,
        float* __restrict__ out, float* __restrict__ ws_f, unsigned* __restrict__ ctrl) {
  extern __shared__ float As[];                 // [KG_][16][4] fragment-major A slab
  const int tid  = threadIdx.x;
  const int wv   = tid >> 5;
  const int lane = tid & 31;
  const int m16  = lane & 15;
  const int half = lane >> 4;                   // A frag: k-offset = 2*half
  const int r    = blockIdx.x & 1;              // row block (batches r*16..r*16+15)
  const int cgrp = blockIdx.x >> 1;             // 8 col-tiles per WG

  float*       hbuf = ws_f + WS_H / 4;
  const float* Wp   = ws_f + WS_WP / 4;
  const float* Rp   = ws_f + WS_RP / 4;

  for (int t = 0; t < T_; ++t) {
    const float* hcur = hbuf + (size_t)(t & 1) * (B_ * H_);
    float*       hnxt = hbuf + (size_t)((t + 1) & 1) * (B_ * H_);

    // ---- stage A = [x_t rows | h_t rows] for this WG's 16 batch rows into LDS ----
    // Async global->LDS, per-lane fragment-major scatter; one asynccnt wait at end.
    for (int i = tid; i < 16 * KGX; i += NTHREADS) {
      const int m = i >> 7, kg = i & (KGX - 1);
      const float* src = x + ((size_t)(r * 16 + m) * T_ + t) * IN_ + kg * 4;
      async_copy16(src, As + (kg * 16 + m) * 4);
      if ((kg & 7) == 0 && t + 1 < T_) __builtin_prefetch(src + IN_, 0, 1);  // next x_t
    }
    for (int i = tid; i < 16 * KGH; i += NTHREADS) {
      const int m = i >> 8, kg = i & (KGH - 1);
      async_copy16(hcur + (size_t)(r * 16 + m) * H_ + kg * 4,
                   As + ((KGX + kg) * 16 + m) * 4);
    }
    async_wait0();
    __syncthreads();

    if (wv < 8) {
      // ---- one 16x16 tile of h_new: D = A(16x1536) x B(1536x16), fp32 WMMA ----
      const int c = cgrp * 8 + wv;
      v8f acc = {0.f, 0.f, 0.f, 0.f, 0.f, 0.f, 0.f, 0.f};
      const float* ap = As + m16 * 4 + half * 2;
      const float* bp = Wp + ((size_t)c * 16 + m16) * 4 + half * 2;
      #pragma unroll 8
      for (int kg = 0; kg < KG_; ++kg) {
        const v2f a = *(const v2f*)(ap + kg * 64);
        const v2f b = *(const v2f*)(bp + (size_t)kg * (H_ * 4));
        acc = __builtin_amdgcn_wmma_f32_16x16x4_f32(false, a, false, b,
                                                    (short)0, acc, false, false);
      }
      // leaky-integrator update; h_old fetched from the staged LDS slab
      const int col = c * 16 + m16;
      #pragma unroll
      for (int i = 0; i < 8; ++i) {
        const int mloc = i + half * 8;          // D layout: vgpr i -> M = i + 8*half
        const float hold = As[((KGX + (col >> 2)) * 16 + mloc) * 4 + (col & 3)];
        const float hnew = (1.0f - ALPHA) * hold + ALPHA * tanhf(acc[i]);
        hnxt[(size_t)(r * 16 + mloc) * H_ + col] = hnew;
      }
    } else if (blockIdx.x < 2 && t > 0) {
      // ---- readout of h_t (published last step): y[:, t-1] = h_t @ r^T + b ----
      v8f acc = {0.f, 0.f, 0.f, 0.f, 0.f, 0.f, 0.f, 0.f};
      const float* ap = As + KGX * 64 + m16 * 4 + half * 2;   // h part of slab
      const float* bp = Rp + m16 * 4 + half * 2;
      #pragma unroll 4
      for (int kg = 0; kg < KGH; ++kg) {
        const v2f a = *(const v2f*)(ap + kg * 64);
        const v2f b = *(const v2f*)(bp + kg * 64);
        acc = __builtin_amdgcn_wmma_f32_16x16x4_f32(false, a, false, b,
                                                    (short)0, acc, false, false);
      }
      const float bias = rbias[m16];
      #pragma unroll
      for (int i = 0; i < 8; ++i) {
        const int bm = r * 16 + i + half * 8;
        out[((size_t)bm * T_ + (t - 1)) * O_ + m16] = acc[i] + bias;
      }
    }

    grid_sync(ctrl, ctrl + 1);   // h_{t+1} visible to all WGPs before next step
  }

  // ---- epilogue: y[:, T-1] from h_T (lives in hbuf[0] since T is even) ----
  if (blockIdx.x < 2) {
    const float* hfin = hbuf;    // (T_ & 1) == 0
    for (int i = tid; i < 16 * KGH; i += NTHREADS) {
      const int m = i >> 8, kg = i & (KGH - 1);
      async_copy16(hfin + (size_t)(r * 16 + m) * H_ + kg * 4,
                   As + ((KGX + kg) * 16 + m) * 4);
    }
    async_wait0();
    __syncthreads();
    if (wv == 8) {
      v8f acc = {0.f, 0.f, 0.f, 0.f, 0.f, 0.f, 0.f, 0.f};
      const float* ap = As + KGX * 64 + m16 * 4 + half * 2;
      const float* bp = Rp + m16 * 4 + half * 2;
      #pragma unroll 4
      for (int kg = 0; kg < KGH; ++kg) {
        const v2f a = *(const v2f*)(ap + kg * 64);
        const v2f b = *(const v2f*)(bp + kg * 64);
        acc = __builtin_amdgcn_wmma_f32_16x16x4_f32(false, a, false, b,
                                                    (short)0, acc, false, false);
      }
      const float bias = rbias[m16];
      #pragma unroll
      for (int i = 0; i < 8; ++i) {
        const int bm = r * 16 + i + half * 8;
        out[((size_t)bm * T_ + (T_ - 1)) * O_ + m16] = acc[i] + bias;
      }
    }
  }
}

extern "C" void kernel_launch(void* const* d_in, const int* in_sizes, int n_in,
                              void* d_out, int out_size, void* d_ws, size_t ws_size,
                              hipStream_t stream) {
  (void)in_sizes; (void)n_in; (void)out_size; (void)ws_size;  // needs ws_size >= ~6.4 MB
  const float* x   = (const float*)d_in[0];
  const float* Win = (const float*)d_in[1];
  const float* W   = (const float*)d_in[2];
  const float* Rw  = (const float*)d_in[3];
  const float* Rb  = (const float*)d_in[4];
  float*    ws_f = (float*)d_ws;
  unsigned* ws_u = (unsigned*)d_ws;

  rc_init<<<dim3(512), dim3(256), 0, stream>>>(Win, W, Rw, ws_f, ws_u);
  rc_main<<<dim3(NWG), dim3(NTHREADS), LDS_BYTES, stream>>>(
      x, Rb, (float*)d_out, ws_f, ws_u);
}